// SVMProjection_35347580846575
// MI455X (gfx1250) — compile-verified
//
#include <hip/hip_runtime.h>
#include <stdint.h>

// out[n,:] = x[n,:] - (x[n,:]·cav - bias·cav) * cav      (N=8192, D=4096, fp32)
// Memory-bound (0.5 FLOP/B): single pass over x, async global->LDS double
// buffering (CDNA5 async engine, ASYNCcnt), ds_swizzle wave32 reductions,
// non-temporal b128 loads/stores so the 256 MiB stream bypasses L2 priority.

typedef float v4f __attribute__((ext_vector_type(4)));

#define TPB 256                 // 8 wave32s per block
#define NWAVES (TPB / 32)
#define ROWS_PER_BLOCK 8
#define DIM 4096
#define DF4 (DIM / 4)           // 1024 float4 per row
#define F4T (DF4 / TPB)         // 4 float4 per thread per row

// ---- CDNA5 async global->LDS copy (16B per lane per issue), NT hint ----
__device__ __forceinline__ void async_f4_to_lds(const v4f* gsrc, const v4f* lds_dst) {
    // low 32 bits of a flat LDS pointer == wave-relative LDS byte address
    unsigned lds = (unsigned)(uintptr_t)lds_dst;
    unsigned long long ga = (unsigned long long)(uintptr_t)gsrc;
    asm volatile("global_load_async_to_lds_b128 %0, %1, off th:TH_LOAD_NT"
                 :: "v"(lds), "v"(ga) : "memory");
}
__device__ __forceinline__ void wait_async_le4() {
    asm volatile("s_wait_asynccnt 4" ::: "memory");
}
__device__ __forceinline__ void wait_async_0() {
    asm volatile("s_wait_asynccnt 0" ::: "memory");
}

// wave32 XOR-swap reduction: one ds_swizzle_b32 per step, no index math.
// Group-of-32 encoding: offset = (xor_mask<<10) | (or<<5) | and_mask(0x1F).
__device__ __forceinline__ float wave_reduce_sum(float v) {
    v += __int_as_float(__builtin_amdgcn_ds_swizzle(__float_as_int(v), 0x401F)); // SWAPX16
    v += __int_as_float(__builtin_amdgcn_ds_swizzle(__float_as_int(v), 0x201F)); // SWAPX8
    v += __int_as_float(__builtin_amdgcn_ds_swizzle(__float_as_int(v), 0x101F)); // SWAPX4
    v += __int_as_float(__builtin_amdgcn_ds_swizzle(__float_as_int(v), 0x081F)); // SWAPX2
    v += __int_as_float(__builtin_amdgcn_ds_swizzle(__float_as_int(v), 0x041F)); // SWAPX1
    return v;
}

// Single-barrier block reduce+broadcast. Caller alternates red8 between two
// buffers per call: call k's reads are fenced from call k+2's writes by call
// k+1's barrier, so no trailing __syncthreads is needed.
__device__ __forceinline__ float block_reduce_bcast(float v, float* red8) {
    v = wave_reduce_sum(v);
    const int t = threadIdx.x;
    if ((t & 31) == 0) red8[t >> 5] = v;
    __syncthreads();
    float s = 0.f;
    #pragma unroll
    for (int w = 0; w < NWAVES; ++w) s += red8[w];
    return s;
}

__global__ __launch_bounds__(TPB) void svm_rank1_reject_kernel(
    const float* __restrict__ x, const float* __restrict__ cav,
    const float* __restrict__ bias, float* __restrict__ out, int nrows)
{
    __shared__ v4f   buf[2][DF4];       // 32 KiB double-buffered row staging
    __shared__ float red[2][NWAVES];    // phase-alternating reduce scratch

    const int t = threadIdx.x;
    const v4f* cav4  = (const v4f*)cav;
    const v4f* bias4 = (const v4f*)bias;

    // cav chunk lives in registers for the whole block; fold bias·cav once.
    v4f c[F4T];
    float bp = 0.f;
    #pragma unroll
    for (int k = 0; k < F4T; ++k) {
        const int j = t + k * TPB;
        c[k] = cav4[j];
        v4f b = bias4[j];
        bp += b.x * c[k].x + b.y * c[k].y + b.z * c[k].z + b.w * c[k].w;
    }
    int phase = 0;
    const float bproj = block_reduce_bcast(bp, red[phase & 1]); ++phase;

    const int row0 = blockIdx.x * ROWS_PER_BLOCK;
    int rows = nrows - row0;
    if (rows <= 0) return;
    if (rows > ROWS_PER_BLOCK) rows = ROWS_PER_BLOCK;

    // stage row 0 into buf[0] via the async engine
    {
        const v4f* src = (const v4f*)(x + (size_t)row0 * DIM);
        #pragma unroll
        for (int k = 0; k < F4T; ++k) {
            const int j = t + k * TPB;
            async_f4_to_lds(&src[j], &buf[0][j]);
        }
    }

    for (int i = 0; i < rows; ++i) {
        const int cur = i & 1;
        if (i + 1 < rows) {
            // prefetch next row into the other buffer, then retire current row
            const v4f* nsrc = (const v4f*)(x + (size_t)(row0 + i + 1) * DIM);
            #pragma unroll
            for (int k = 0; k < F4T; ++k) {
                const int j = t + k * TPB;
                async_f4_to_lds(&nsrc[j], &buf[cur ^ 1][j]);
            }
            wait_async_le4();   // async loads complete in order: oldest 4 (row i) done
        } else {
            wait_async_0();
        }

        // dot pass: LDS -> regs (kept for the update pass)
        v4f xr[F4T];
        float p = 0.f;
        #pragma unroll
        for (int k = 0; k < F4T; ++k) {
            xr[k] = buf[cur][t + k * TPB];
            p += xr[k].x * c[k].x + xr[k].y * c[k].y + xr[k].z * c[k].z + xr[k].w * c[k].w;
        }
        const float s = block_reduce_bcast(p, red[phase & 1]) - bproj; ++phase;

        // update pass: streaming non-temporal b128 stores
        v4f* out4 = (v4f*)(out + (size_t)(row0 + i) * DIM);
        #pragma unroll
        for (int k = 0; k < F4T; ++k) {
            const int j = t + k * TPB;
            v4f o;
            o.x = xr[k].x - s * c[k].x;
            o.y = xr[k].y - s * c[k].y;
            o.z = xr[k].z - s * c[k].z;
            o.w = xr[k].w - s * c[k].w;
            __builtin_nontemporal_store(o, &out4[j]);
        }
    }
}

extern "C" void kernel_launch(void* const* d_in, const int* in_sizes, int n_in,
                              void* d_out, int out_size, void* d_ws, size_t ws_size,
                              hipStream_t stream) {
    const float* x    = (const float*)d_in[0];
    const float* cav  = (const float*)d_in[1];
    const float* bias = (const float*)d_in[2];
    float* out = (float*)d_out;

    const int d = in_sizes[1];               // 4096 (kernel compiled for DIM)
    const int n = in_sizes[0] / d;           // 8192 rows
    const int grid = (n + ROWS_PER_BLOCK - 1) / ROWS_PER_BLOCK;

    svm_rank1_reject_kernel<<<grid, TPB, 0, stream>>>(x, cav, bias, out, n);
}